// GCNLayer_68839735821095
// MI455X (gfx1250) — compile-verified
//
#include <hip/hip_runtime.h>

#define D_FEAT 256
#define SCAN_THREADS 1024

typedef __attribute__((ext_vector_type(2))) float v2f;
typedef __attribute__((ext_vector_type(8))) float v8f;

// ---------------------------------------------------------------- K1: zero deg + cnt
__global__ void gcn_init(float* __restrict__ deg, int* __restrict__ cnt, int n) {
    int i = blockIdx.x * blockDim.x + threadIdx.x;
    if (i < n) { deg[i] = 0.0f; cnt[i] = 0; }
}

// ---------------------------------------------------------------- K2: count (deg by col, cnt by row)
__global__ void gcn_count(const int* __restrict__ row, const int* __restrict__ col,
                          float* __restrict__ deg, int* __restrict__ cnt, int E) {
    int e = blockIdx.x * blockDim.x + threadIdx.x;
    if (e < E) {
        unsafeAtomicAdd(&deg[col[e]], 1.0f);   // global_atomic_add_f32 (no return)
        atomicAdd(&cnt[row[e]], 1);            // global_atomic_add_u32
    }
}

// ---------------------------------------------------------------- K3: dinv
__global__ void gcn_dinv(float* __restrict__ deg, int n) {
    int i = blockIdx.x * blockDim.x + threadIdx.x;
    if (i < n) deg[i] = rsqrtf(1.0f + deg[i]);
}

// ---------------------------------------------------------------- K4: exclusive scan (1 block)
// start[0..N] = exclusive prefix of cnt; fill[i] = start[i] (running cursor for placement)
__global__ void gcn_scan(const int* __restrict__ cnt, int* __restrict__ start,
                         int* __restrict__ fill, int N) {
    __shared__ int partial[SCAN_THREADS];
    const int t = threadIdx.x;
    const int chunk = (N + SCAN_THREADS - 1) / SCAN_THREADS;
    const int lo = t * chunk;
    const int hi = (lo + chunk < N) ? lo + chunk : N;

    int s = 0;
    for (int i = lo; i < hi; ++i) s += cnt[i];
    partial[t] = s;
    __syncthreads();

    if (t == 0) {
        int run = 0;
        for (int i = 0; i < SCAN_THREADS; ++i) { int v = partial[i]; partial[i] = run; run += v; }
        start[N] = run;   // total edge count
    }
    __syncthreads();

    int run = partial[t];
    for (int i = lo; i < hi; ++i) {
        start[i] = run;
        fill[i]  = run;
        run += cnt[i];
    }
}

// ---------------------------------------------------------------- K5: CSR placement
__global__ void gcn_place(const int* __restrict__ row, const int* __restrict__ col,
                          int* __restrict__ fill, int* __restrict__ eidx, int E) {
    int e = blockIdx.x * blockDim.x + threadIdx.x;
    if (e < E) {
        int pos = atomicAdd(&fill[row[e]], 1);
        eidx[pos] = col[e];
    }
}

// ---------------------------------------------------------------- K6: WMMA GEMM
// scaled[m][n] = (sum_k X[m][k]*W[k][n] + b[n]) * dinv[m]
// One wave per 16x16 tile via V_WMMA_F32_16X16X4_F32, K stepped by 4.
__global__ void gcn_gemm_scale(const float* __restrict__ X,
                               const float* __restrict__ W,
                               const float* __restrict__ bias,
                               const float* __restrict__ dinv,
                               float* __restrict__ scaled,
                               int N) {
    const int lane = threadIdx.x & 31;
    const int wid  = threadIdx.x >> 5;
    const int tile = blockIdx.x * (blockDim.x >> 5) + wid;

    const int ntn = D_FEAT / 16;
    const int tm  = tile / ntn;
    const int tn  = tile % ntn;
    if (tm * 16 >= N) return;             // whole wave exits together (EXEC stays all-1s)

    const int m0   = tm * 16;
    const int n0   = tn * 16;
    const int half = lane >> 4;
    const int l15  = lane & 15;
    const int mA   = m0 + l15;            // A-matrix row for this lane
    const int n    = n0 + l15;            // B/C/D column for this lane

    v8f acc = {};

#if __has_builtin(__builtin_amdgcn_wmma_f32_16x16x4_f32)
    for (int k = 0; k < D_FEAT; k += 4) {
        // A 16x4: lanes 0-15 hold K={k,k+1}; lanes 16-31 hold K={k+2,k+3}
        const float* ap = X + (size_t)mA * D_FEAT + k + 2 * half;
        v2f a; a.x = ap[0]; a.y = ap[1];
        // B 4x16: lanes 0-15 rows K={k,k+1}; lanes 16-31 rows K={k+2,k+3}
        const float* bp = W + (size_t)(k + 2 * half) * D_FEAT + n;
        v2f bb; bb.x = bp[0]; bb.y = bp[D_FEAT];
        acc = __builtin_amdgcn_wmma_f32_16x16x4_f32(
            false, a, false, bb, (short)0, acc, false, false);
    }
#else
    for (int r = 0; r < 8; ++r) {
        int mm = m0 + r + 8 * half;
        float s = 0.0f;
        for (int k = 0; k < D_FEAT; ++k) s = fmaf(X[(size_t)mm * D_FEAT + k], W[(size_t)k * D_FEAT + n], s);
        acc[r] = s;
    }
#endif

    // C/D: VGPR r -> rows M=r (lanes 0-15) / M=r+8 (lanes 16-31), N = lane&15
    const float bn = bias[n];
#pragma unroll
    for (int r = 0; r < 8; ++r) {
        const int mm = m0 + r + 8 * half;
        scaled[(size_t)mm * D_FEAT + n] = (acc[r] + bn) * dinv[mm];
    }
}

// ---------------------------------------------------------------- K7: CSR gather (no fp atomics)
// out[r] = (scaled[r] + sum_{j in row r} scaled[eidx[j]]) * dinv[r]
// One wave per node; node id forced wave-uniform -> start/eidx are scalar loads;
// per-edge data moves as 2x b128 per lane (full 1KB row per wave), L2-resident.
__global__ void gcn_gather(const int* __restrict__ start,
                           const int* __restrict__ eidx,
                           const float* __restrict__ scaled,
                           const float* __restrict__ dinv,
                           float* __restrict__ out, int N) {
    int r = __builtin_amdgcn_readfirstlane(blockIdx.x * (blockDim.x >> 5) + (threadIdx.x >> 5));
    if (r >= N) return;
    const int lane = threadIdx.x & 31;

    const float4* self = (const float4*)(scaled + (size_t)r * D_FEAT);
    float4 a0 = self[lane];          // d = lane*4 .. lane*4+3        (0..127)
    float4 a1 = self[lane + 32];     // d = 128 + lane*4 ..           (128..255)

    const int s0 = start[r];
    const int s1 = start[r + 1];
    for (int j = s0; j < s1; ++j) {
        const int c = eidx[j];       // wave-uniform -> s_load
        const float4* src = (const float4*)(scaled + (size_t)c * D_FEAT);
        float4 v0 = src[lane];
        float4 v1 = src[lane + 32];
        a0.x += v0.x; a0.y += v0.y; a0.z += v0.z; a0.w += v0.w;
        a1.x += v1.x; a1.y += v1.y; a1.z += v1.z; a1.w += v1.w;
    }

    const float dv = dinv[r];
    a0.x *= dv; a0.y *= dv; a0.z *= dv; a0.w *= dv;
    a1.x *= dv; a1.y *= dv; a1.z *= dv; a1.w *= dv;

    float4* dst = (float4*)(out + (size_t)r * D_FEAT);
    dst[lane]      = a0;
    dst[lane + 32] = a1;
}

// ----------------------------------------------------------------
static inline size_t align1k(size_t x) { return (x + 1023) & ~(size_t)1023; }

extern "C" void kernel_launch(void* const* d_in, const int* in_sizes, int n_in,
                              void* d_out, int out_size, void* d_ws, size_t ws_size,
                              hipStream_t stream) {
    (void)n_in; (void)out_size; (void)ws_size;

    const int*   row  = (const int*)d_in[0];
    const int*   col  = (const int*)d_in[1];
    const float* X    = (const float*)d_in[2];
    const float* W    = (const float*)d_in[3];
    const float* bias = (const float*)d_in[4];
    float*       out  = (float*)d_out;

    const int E = in_sizes[0];
    const int N = in_sizes[2] / D_FEAT;

    // Workspace layout (1KB-aligned sections):
    char* ws = (char*)d_ws;
    size_t off = 0;
    float* deg   = (float*)(ws + off); off = align1k(off + (size_t)N * 4);
    int*   cnt   = (int*)(ws + off);   off = align1k(off + (size_t)N * 4);
    int*   start = (int*)(ws + off);   off = align1k(off + (size_t)(N + 1) * 4);
    int*   fill  = (int*)(ws + off);   off = align1k(off + (size_t)N * 4);
    int*   eidx  = (int*)(ws + off);   off = align1k(off + (size_t)E * 4);
    float* scaled= (float*)(ws + off); // N*256 floats

    gcn_init<<<(N + 255) / 256, 256, 0, stream>>>(deg, cnt, N);
    gcn_count<<<(E + 255) / 256, 256, 0, stream>>>(row, col, deg, cnt, E);
    gcn_dinv<<<(N + 255) / 256, 256, 0, stream>>>(deg, N);
    gcn_scan<<<1, SCAN_THREADS, 0, stream>>>(cnt, start, fill, N);
    gcn_place<<<(E + 255) / 256, 256, 0, stream>>>(row, col, fill, eidx, E);

    const int tiles = ((N + 15) / 16) * (D_FEAT / 16);
    gcn_gemm_scale<<<(tiles + 3) / 4, 128, 0, stream>>>(X, W, bias, deg, scaled, N);

    gcn_gather<<<(N + 7) / 8, 256, 0, stream>>>(start, eidx, scaled, deg, out, N);
}